// MaskedCELoss_412316860847
// MI455X (gfx1250) — compile-verified
//
#include <hip/hip_runtime.h>

// Hierarchical masked CE loss — MI455X (gfx1250), bandwidth-bound streaming kernel.
// ~286 MB of traffic -> ~12us at 23.3 TB/s roofline. Uses async-to-LDS (ASYNCcnt)
// staging overlapped with compute, global_prefetch_b8, NT B128 loads/stores, and
// wave32 ds_swizzle XOR reductions (no address-setup VALU).

#define N0v     128
#define B1v     16
#define B2v     16
#define N1v     2048
#define N2v     32768
#define TOTALN  34944      // N0 + N1 + N2
#define BATCHN  1024
#define NEGV    (-100000000.0f)
#define NTHR    256
#define NWAVES  (NTHR / 32)
#define NINF    (-3.0e38f)   // finite sentinel keeps (m - M) NaN-free

typedef float v4f __attribute__((ext_vector_type(4)));
typedef int   v4i __attribute__((ext_vector_type(4)));
typedef __attribute__((address_space(1))) v4i* gv4i_p;
typedef __attribute__((address_space(3))) v4i* lv4i_p;

#if defined(__gfx1250__) && __has_builtin(__builtin_amdgcn_global_load_async_to_lds_b128)
#define HAVE_ASYNC_LDS 1
#else
#define HAVE_ASYNC_LDS 0
#endif

__device__ __forceinline__ void wait_asynccnt0() {
#if __has_builtin(__builtin_amdgcn_s_wait_asynccnt)
  __builtin_amdgcn_s_wait_asynccnt(0);
#else
  asm volatile("s_wait_asynccnt 0" ::: "memory");
#endif
}

// ds_swizzle group-of-32 XOR patterns: offset = {xor[14:10], or[9:5], and[4:0]}
#define SWZ_X16 0x401f
#define SWZ_X8  0x201f
#define SWZ_X4  0x101f
#define SWZ_X2  0x081f
#define SWZ_X1  0x041f
#define SWZF(v, m) __int_as_float(__builtin_amdgcn_ds_swizzle(__float_as_int(v), (m)))

__device__ __forceinline__ float wave_max_f(float v) {
  v = fmaxf(v, SWZF(v, SWZ_X16));
  v = fmaxf(v, SWZF(v, SWZ_X8));
  v = fmaxf(v, SWZF(v, SWZ_X4));
  v = fmaxf(v, SWZF(v, SWZ_X2));
  v = fmaxf(v, SWZF(v, SWZ_X1));
  return v;
}

__device__ __forceinline__ float wave_sum_f(float v) {
  v += SWZF(v, SWZ_X16);
  v += SWZF(v, SWZ_X8);
  v += SWZF(v, SWZ_X4);
  v += SWZF(v, SWZ_X2);
  v += SWZF(v, SWZ_X1);
  return v;
}

// wave argmax (value, index) with first-index tie-break
#define AM_STEP(mask)                                              \
  { const float ov = SWZF(v, mask);                                \
    const int   oi = __builtin_amdgcn_ds_swizzle(i, mask);         \
    if (ov > v || (ov == v && oi < i)) { v = ov; i = oi; } }
__device__ __forceinline__ void wave_argmax_f(float& v, int& i) {
  AM_STEP(SWZ_X16) AM_STEP(SWZ_X8) AM_STEP(SWZ_X4) AM_STEP(SWZ_X2) AM_STEP(SWZ_X1)
}

// Block logsumexp from per-thread online partials (m, s): returns M + log(S).
// max-pass + single rescale exp + sum-pass. 3 barriers, cheap combines.
__device__ __forceinline__ float block_lse(float m, float s,
                                           float* sm_m, float* sm_s, int tid) {
  const int lane = tid & 31, wid = tid >> 5;
  const float wm = wave_max_f(m);
  if (lane == 0) sm_m[wid] = wm;
  __syncthreads();
  float M = sm_m[0];
  #pragma unroll
  for (int w = 1; w < NWAVES; ++w) M = fmaxf(M, sm_m[w]);
  float ss = wave_sum_f(s * __expf(m - M));
  if (lane == 0) sm_s[wid] = ss;
  __syncthreads();
  float S = sm_s[0];
  #pragma unroll
  for (int w = 1; w < NWAVES; ++w) S += sm_s[w];
  __syncthreads();   // safe reuse of sm_m/sm_s by later reductions
  return M + __logf(S);
}

__global__ __launch_bounds__(NTHR)
void hier_ce_row_kernel(const float* __restrict__ outputs,
                        const int* __restrict__ labels,
                        float* __restrict__ out,
                        float* __restrict__ row_loss) {
  const int tid  = threadIdx.x;
  const int lane = tid & 31, wid = tid >> 5;
  const int r    = blockIdx.x;
  const float* row  = outputs + (size_t)r * TOTALN;
  float*       orow = out     + (size_t)r * TOTALN;

  const int lab0 = labels[r * 3 + 0];
  const int lab1 = labels[r * 3 + 1];
  const int r1   = lab1 - lab0 * B1v;

  __shared__ __align__(16) float sm1[N1v];   // staged out1 tile (8 KB)
  __shared__ float sm_m[NWAVES];
  __shared__ float sm_s[NWAVES];
  __shared__ int   sm_i[NWAVES];
  __shared__ int   bc_pred1;
  __shared__ int   bc_found1;
  __shared__ float bc_loss01;

  // ---- Kick off async DMA of out1 into LDS first: overlaps all of phase A ----
  const float* row1  = row  + N0v;
  float*       orow1 = orow + N0v;
  #pragma unroll
  for (int k = 0; k < (N1v / 4) / NTHR; ++k) {   // 2 x B128 per lane
    const int idx4 = tid + k * NTHR;
#if HAVE_ASYNC_LDS
    __builtin_amdgcn_global_load_async_to_lds_b128(
        (gv4i_p)(row1 + idx4 * 4),
        (lv4i_p)&sm1[idx4 * 4],
        0, 0);
#else
    ((v4f*)sm1)[idx4] = ((const v4f*)row1)[idx4];
#endif
  }

  // ---------------- Phase A: out0 (128): argmax + logsumexp ----------------
  float x0 = NINF; int i0 = 0x7fffffff;
  if (tid < N0v) {
    x0 = row[tid];
    i0 = tid;
    orow[tid] = x0;                  // pass-through copy
  }
  float av = x0; int ai = i0;
  wave_argmax_f(av, ai);
  if (lane == 0) { sm_m[wid] = av; sm_i[wid] = ai; }
  __syncthreads();
  float M0 = sm_m[0]; int P0 = sm_i[0];
  #pragma unroll
  for (int w = 1; w < NWAVES; ++w) {
    const float mv = sm_m[w]; const int mi = sm_i[w];
    if (mv > M0 || (mv == M0 && mi < P0)) { M0 = mv; P0 = mi; }
  }
  float e0 = (tid < N0v) ? __expf(x0 - M0) : 0.0f;
  e0 = wave_sum_f(e0);
  if (lane == 0) sm_s[wid] = e0;
  __syncthreads();
  float S0 = sm_s[0];
  #pragma unroll
  for (int w = 1; w < NWAVES; ++w) S0 += sm_s[w];
  const int   pred0 = P0;
  const float lse0  = M0 + __logf(S0);
  __syncthreads();   // sm_* free for reuse

  // ---------------- Phase B: out1 (2048) ----------------
#if HAVE_ASYNC_LDS
  wait_asynccnt0();
#endif
  __syncthreads();   // all waves' DMA complete, sm1 visible

  // Fused single pass over sm1: online logsumexp + masked NT writeback
  // (mask depends only on pred0, already known).
  const v4f negv4 = {NEGV, NEGV, NEGV, NEGV};
  float mB = NINF, sB = 0.0f;
  #pragma unroll
  for (int k = 0; k < (N1v / 4) / NTHR; ++k) {   // 2 x v4f per lane
    const int idx4 = tid + k * NTHR;
    const v4f v = ((const v4f*)sm1)[idx4];
    const float mm = fmaxf(fmaxf(v.x, v.y), fmaxf(v.z, v.w));
    if (mm > mB) { sB *= __expf(mB - mm); mB = mm; }
    sB += __expf(v.x - mB) + __expf(v.y - mB) + __expf(v.z - mB) + __expf(v.w - mB);
    v4f w = v;
    if ((idx4 >> 2) != pred0) w = negv4;
    __builtin_nontemporal_store(w, ((v4f*)orow1) + idx4);
  }
  const float lse1_full = block_lse(mB, sB, sm_m, sm_s, tid);

  if (tid == 0) {
    const bool found0 = (pred0 != lab0);
    float gm = NINF;
    for (int k = 0; k < B1v; ++k) gm = fmaxf(gm, sm1[lab0 * B1v + k]);
    float gs = 0.0f;
    for (int k = 0; k < B1v; ++k) gs += __expf(sm1[lab0 * B1v + k] - gm);
    const float lse1_grp = gm + __logf(gs);
    int p1 = 0; float bv = sm1[pred0 * B1v];
    for (int k = 1; k < B1v; ++k) {
      const float vv = sm1[pred0 * B1v + k];
      if (vv > bv) { bv = vv; p1 = k; }
    }
    // out1[lab1] == out1[lab0*16 + r1] serves both CE branches
    const float loss1 = (found0 ? lse1_full : lse1_grp) - sm1[lab1];
    const float loss0 = lse0 - row[lab0];
    bc_pred1  = pred0 * B1v + p1;
    bc_found1 = (found0 || (p1 != r1)) ? 1 : 0;
    bc_loss01 = loss0 + loss1;
  }
  __syncthreads();
  const int pred1 = bc_pred1;

  // ---------------- Phase C: out2 (32768), single streamed pass ----------------
  const float* row2   = row  + N0v + N1v;
  float*       orow2  = orow + N0v + N1v;
  const v4f*   row2v  = (const v4f*)row2;
  v4f*         orow2v = (v4f*)orow2;

  float mC = NINF, sC = 0.0f;
  #pragma unroll 4
  for (int k = 0; k < (N2v / 4) / NTHR; ++k) {   // 32 x B128 per lane
    const int idx4 = tid + k * NTHR;
    if ((k & 3) == 0)   // one prefetch per unrolled group of 4
      __builtin_prefetch(row2 + (size_t)(idx4 + 8 * NTHR) * 4, 0, 0); // global_prefetch_b8
    v4f v = __builtin_nontemporal_load(row2v + idx4);
    const float mm = fmaxf(fmaxf(v.x, v.y), fmaxf(v.z, v.w));
    if (mm > mC) { sC *= __expf(mC - mm); mC = mm; }   // online rescale
    sC += __expf(v.x - mC) + __expf(v.y - mC) + __expf(v.z - mC) + __expf(v.w - mC);
    v4f w = v;
    if ((idx4 >> 2) != pred1) w = negv4;
    __builtin_nontemporal_store(w, orow2v + idx4);
  }
  const float lse2_full = block_lse(mC, sC, sm_m, sm_s, tid);

  if (tid == 0) {
    float gm = NINF;
    for (int k = 0; k < B2v; ++k) gm = fmaxf(gm, row2[lab1 * B2v + k]);
    float gs = 0.0f;
    for (int k = 0; k < B2v; ++k) gs += __expf(row2[lab1 * B2v + k] - gm);
    const float lse2_grp = gm + __logf(gs);
    // out2[lab2] == out2[lab1*16 + r2] serves both branches
    const int lab2 = labels[r * 3 + 2];
    const float loss2 = (bc_found1 ? lse2_full : lse2_grp) - row2[lab2];
    row_loss[r] = bc_loss01 + loss2;
  }
}

// Deterministic fixed-order sum of per-row losses (no atomics).
__global__ __launch_bounds__(256)
void loss_sum_kernel(const float* __restrict__ row_loss, float* __restrict__ out_loss) {
  __shared__ float red[256];
  const int tid = threadIdx.x;
  float s = 0.0f;
  #pragma unroll
  for (int k = 0; k < BATCHN / 256; ++k) s += row_loss[tid + k * 256];
  red[tid] = s;
  __syncthreads();
  for (int st = 128; st > 0; st >>= 1) {
    if (tid < st) red[tid] += red[tid + st];
    __syncthreads();
  }
  if (tid == 0) *out_loss = red[0];
}

extern "C" void kernel_launch(void* const* d_in, const int* in_sizes, int n_in,
                              void* d_out, int out_size, void* d_ws, size_t ws_size,
                              hipStream_t stream) {
  const float* outputs = (const float*)d_in[0];   // (1024, 34944) f32
  const int*   labels  = (const int*)d_in[1];     // (1024, 3) i32
  float* out      = (float*)d_out;                // 1024*34944 floats + 1 loss
  float* row_loss = (float*)d_ws;                 // 1024 floats of scratch

  hier_ce_row_kernel<<<BATCHN, NTHR, 0, stream>>>(outputs, labels, out, row_loss);
  loss_sum_kernel<<<1, 256, 0, stream>>>(row_loss, out + (size_t)BATCHN * TOTALN);
}